// Node_15401752723588
// MI455X (gfx1250) — compile-verified
//
#include <hip/hip_runtime.h>

// ---------------------------------------------------------------------------
// MI455X (gfx1250) implementation.
//
// Cost model: output zero-fill = 256 MB stores -> ~11us floor @ 23.3 TB/s.
// Everything else (512KB gather, 0.5 MFLOP of GEMM, 2-channel batchnorm) is
// noise. Strategy: wide non-temporal b128 stores for the fill, WMMA
// (v_wmma_f32_16x16x4_f32) for the tiny GEMM at zero extra memory cost, and
// a fixed-order (deterministic, atomic-free) reduction for mean/var.
//
// Round-2 fix: W1 B-fragments are now built with unconditional clamped loads
// + value-level select (v_cndmask) instead of exec-masked branches, and are
// hoisted out of the K-loop, so the WMMA loop is branch-free with EXEC=all-1s
// throughout (required by the WMMA spec, and removes s_and_saveexec churn).
// ---------------------------------------------------------------------------

typedef float v2f __attribute__((ext_vector_type(2)));
typedef float v4f __attribute__((ext_vector_type(4)));
typedef float v8f __attribute__((ext_vector_type(8)));

constexpr int   B_ROWS = 4096;
constexpr int   N_COLS = 16384;
constexpr int   IN_CH  = 32;
constexpr int   HID    = 2;
constexpr int   OUT_CH = 16;
constexpr float EPS    = 1e-5f;

constexpr int ROWS_PER_WAVE = 16;                      // one 16x16 WMMA tile
constexpr int NWAVES        = B_ROWS / ROWS_PER_WAVE;  // 256 waves total
constexpr int KSTEPS        = IN_CH / 4;               // 8 WMMA K-steps

// Workspace layout (floats):
//   [0,                B*HID)         h  (pre-norm hidden, 4096x2)
//   [B*HID,            +NWAVES*8)     per-wave partials {sum,sumsq} x {g,n}
//   [B*HID+NWAVES*8,   +4)            stats: sum0,sum1, sumsq0,sumsq1
constexpr int WS_H        = 0;
constexpr int WS_PARTIAL  = B_ROWS * HID;
constexpr int WS_STATS    = WS_PARTIAL + NWAVES * 8;

// ---------------------------------------------------------------------------
// Kernel 1: h = gather(x, in_ixs) @ W1 + b1 via v_wmma_f32_16x16x4_f32.
// One wave handles 16 rows. A-fragment layout (ISA 7.12.2, 32-bit A 16x4):
//   lanes 0-15  : M=lane,    v0=K0, v1=K1
//   lanes 16-31 : M=lane-16, v0=K2, v1=K3
// B-fragment (4x16, striped like C/D):
//   v0: lanes 0-15 K=0, lanes 16-31 K=2;  v1: K=1 / K=3;  N=lane%16.
// K=32 -> 8 accumulating WMMA steps. W1 (32x2) is zero-padded to N=16.
// ---------------------------------------------------------------------------
__global__ __launch_bounds__(256) void k_gemm1(const float* __restrict__ x,
                                               const int*   __restrict__ in_ixs,
                                               const float* __restrict__ W1,
                                               const float* __restrict__ b1,
                                               float*       __restrict__ ws)
{
    float* h       = ws + WS_H;
    float* partial = ws + WS_PARTIAL;

    const int lane = threadIdx.x & 31;
    const int wave = (blockIdx.x * blockDim.x + threadIdx.x) >> 5;  // 0..255
    const int row0 = wave * ROWS_PER_WAVE;

    const int  m = lane & 15;          // row within tile / N within B-frag
    const int  g = lane >> 4;          // half-wave group
    const long rowbase = (long)(row0 + m) * N_COLS;

    // Hoist all B-fragments (W1 zero-padded from 4x2 to 4x16, per K-step).
    // Unconditional loads from a clamped address + value select -> no
    // divergent branches, EXEC stays all-1s for the WMMA loop.
    const int  nc    = (m < HID) ? m : (HID - 1);   // clamped column
    const bool live  = (m < HID);
    v2f bfrag[KSTEPS];
#pragma unroll
    for (int s = 0; s < KSTEPS; ++s) {
        const int k0 = 4 * s + g * 2;
        const float w0 = W1[(k0    ) * HID + nc];
        const float w1 = W1[(k0 + 1) * HID + nc];
        bfrag[s].x = live ? w0 : 0.0f;              // v_cndmask, not a branch
        bfrag[s].y = live ? w1 : 0.0f;
    }

    v8f acc = {};  // C/D accumulator: 16x16 f32, 8 VGPRs
#pragma unroll
    for (int s = 0; s < KSTEPS; ++s) {
        const int k0 = 4 * s + g * 2;  // this lane's K for fragment element 0
        v2f a;
        a.x = x[rowbase + (long)in_ixs[k0]];
        a.y = x[rowbase + (long)in_ixs[k0 + 1]];

        // 8-arg form: (neg_a, A, neg_b, B, c_mod, C, reuse_a, reuse_b)
        acc = __builtin_amdgcn_wmma_f32_16x16x4_f32(
            false, a, false, bfrag[s], (short)0, acc, false, false);
    }

    // Extract D: lane holds column N=m for rows M = v + 8*g (v = VGPR index).
    // Only N<HID columns carry real data; they also build per-wave partials.
    if (m < HID) {
        const int   n    = m;
        const float bias = b1[n];
        float sum = 0.0f, ssum = 0.0f;
#pragma unroll
        for (int v = 0; v < 8; ++v) {
            const int   r  = row0 + v + 8 * g;
            const float hv = acc[v] + bias;
            h[r * HID + n] = hv;
            sum  += hv;
            ssum += hv * hv;
        }
        const int q = g * 2 + n;                 // 0..3 within this wave
        partial[wave * 8 + q * 2 + 0] = sum;
        partial[wave * 8 + q * 2 + 1] = ssum;
    }
}

// ---------------------------------------------------------------------------
// Kernel 2: fixed-order reduction of per-wave partials -> sums / sum-squares.
// One wave; 4 active threads; deterministic accumulation order.
// ---------------------------------------------------------------------------
__global__ __launch_bounds__(32) void k_stats(float* __restrict__ ws)
{
    const float* partial = ws + WS_PARTIAL;
    float*       stats   = ws + WS_STATS;
    const int t = threadIdx.x;
    if (t < 4) {
        const int n    = t & 1;   // channel
        const int isSq = t >> 1;  // 0: sum, 1: sumsq
        float s = 0.0f;
        for (int w = 0; w < NWAVES; ++w)
            for (int g = 0; g < 2; ++g)
                s += partial[w * 8 + (g * 2 + n) * 2 + isSq];
        stats[t] = s;             // stats = {sum0, sum1, ss0, ss1}
    }
}

// ---------------------------------------------------------------------------
// Kernel 3: per output row -- batchnorm + ELU + (h @ W2 + b2) + tanh for the
// 16 live columns; zero-fill the whole 64KB row with non-temporal b128
// stores (output is 256MB > 192MB L2, write-once -> NT avoids pollution);
// barrier; overwrite the 16 scatter positions. One block per row.
// ---------------------------------------------------------------------------
__global__ __launch_bounds__(256) void k_fill_out(const float* __restrict__ ws,
                                                  const int*   __restrict__ out_ixs,
                                                  const float* __restrict__ W2,
                                                  const float* __restrict__ b2,
                                                  const float* __restrict__ gamma,
                                                  const float* __restrict__ beta,
                                                  float*       __restrict__ out)
{
    const int b   = blockIdx.x;
    const int tid = threadIdx.x;

    __shared__ float sy[OUT_CH];
    if (tid < OUT_CH) {
        const float* h     = ws + WS_H;
        const float* stats = ws + WS_STATS;
        float hn[HID];
#pragma unroll
        for (int c = 0; c < HID; ++c) {
            const float mean = stats[c]     * (1.0f / B_ROWS);
            const float var  = stats[2 + c] * (1.0f / B_ROWS) - mean * mean;
            float v = (h[b * HID + c] - mean) * rsqrtf(var + EPS) * gamma[c]
                      + beta[c];
            hn[c] = (v > 0.0f) ? v : expm1f(v);          // ELU(alpha=1)
        }
        float acc = b2[tid];
#pragma unroll
        for (int c = 0; c < HID; ++c) acc += hn[c] * W2[c * OUT_CH + tid];
        sy[tid] = tanhf(acc);
    }

    // Zero-fill this row: 16384 floats = 4096 b128 stores / 256 threads.
    float*    rowp = out + (size_t)b * N_COLS;
    const v4f z    = {0.0f, 0.0f, 0.0f, 0.0f};
#pragma unroll
    for (int i = 0; i < N_COLS / 4 / 256; ++i) {         // 16 iterations
        __builtin_nontemporal_store(z, (v4f*)(rowp + (size_t)(i * 256 + tid) * 4));
    }

    __syncthreads();   // order the zero stores before the scatter overwrites

    if (tid < OUT_CH) {
        rowp[out_ixs[tid]] = sy[tid];
    }
}

// ---------------------------------------------------------------------------
// Inputs (setup_inputs order): x, in_ixs, out_ixs, W1, b1, gamma, beta, W2, b2
// ---------------------------------------------------------------------------
extern "C" void kernel_launch(void* const* d_in, const int* in_sizes, int n_in,
                              void* d_out, int out_size, void* d_ws, size_t ws_size,
                              hipStream_t stream)
{
    const float* x       = (const float*)d_in[0];
    const int*   in_ixs  = (const int*)  d_in[1];
    const int*   out_ixs = (const int*)  d_in[2];
    const float* W1      = (const float*)d_in[3];
    const float* b1      = (const float*)d_in[4];
    const float* gamma   = (const float*)d_in[5];
    const float* beta    = (const float*)d_in[6];
    const float* W2      = (const float*)d_in[7];
    const float* b2      = (const float*)d_in[8];

    float* ws  = (float*)d_ws;
    float* out = (float*)d_out;

    // 256 waves total -> 32 blocks x 256 threads (8 waves/block).
    k_gemm1   <<<NWAVES / 8, 256, 0, stream>>>(x, in_ixs, W1, b1, ws);
    k_stats   <<<1,          32,  0, stream>>>(ws);
    k_fill_out<<<B_ROWS,     256, 0, stream>>>(ws, out_ixs, W2, b2, gamma, beta, out);
}